// IRVLayer_76914274337445
// MI455X (gfx1250) — compile-verified
//
#include <hip/hip_runtime.h>

typedef float v2f __attribute__((ext_vector_type(2)));
typedef float v8f __attribute__((ext_vector_type(8)));

// Problem constants (from reference): B=65536, NT=24, K=64.
constexpr int KDIM = 64;
constexpr int ROW_STRIDE = 2 * KDIM;      // 128 floats per row (64 sim + 64 ys)
constexpr int ROWS_PER_WAVE = 16;         // one 16x16 WMMA tile of rows per wave
constexpr int WAVES_PER_BLOCK = 8;        // 256 threads = 8 wave32

__global__ __launch_bounds__(256)
void irv_wmma_kernel(const float* __restrict__ inputs,
                     const float* __restrict__ V,
                     const float* __restrict__ W,
                     const float* __restrict__ b,
                     const float* __restrict__ b2,
                     float* __restrict__ out,
                     long long totalRows)
{
    const int lane = threadIdx.x & 31;
    const long long wave =
        ((long long)blockIdx.x * blockDim.x + threadIdx.x) >> 5;
    const long long rowBase = wave * ROWS_PER_WAVE;
    if (rowBase >= totalRows) return;           // wave-uniform guard (EXEC stays full)

    // A-matrix layout for V_WMMA_F32_16X16X4_F32:
    //   lanes 0-15  : row = lane,    K = {k0, k0+1}
    //   lanes 16-31 : row = lane-16, K = {k0+2, k0+3}
    const int rowInTile = lane & 15;
    const int koff      = (lane >> 4) << 1;     // 0 or 2

    const float V0 = V[0], V1 = V[1];
    const float W0 = W[0], W1 = W[1];
    const float bb = b[0], bb2 = b2[0];

    const float* rowPtr =
        inputs + (rowBase + rowInTile) * (long long)ROW_STRIDE;

    v8f c = {};                 // f32 accumulator tile (row sums)
    v2f ones; ones.x = 1.0f; ones.y = 1.0f;   // all-ones B => D[m][n] = sum_k A[m][k]

#pragma unroll
    for (int i = 0; i < 16; ++i) {
        const int k0 = i * 4 + koff;
        const v2f s = *(const v2f*)(rowPtr + k0);          // sim pair
        const v2f y = *(const v2f*)(rowPtr + KDIM + k0);   // ys pair (0.0 / 1.0)

        // R = sigmoid(b + W0*sim + W1*rank), rank = k+1
        const float x0 = bb + W0 * s.x + W1 * (float)(k0 + 1);
        const float x1 = bb + W0 * s.y + W1 * (float)(k0 + 2);
        const float r0 = __builtin_amdgcn_rcpf(1.0f + __expf(-x0));
        const float r1 = __builtin_amdgcn_rcpf(1.0f + __expf(-x1));
        const float v0 = (y.x > 0.5f) ? V1 : V0;
        const float v1 = (y.y > 0.5f) ? V1 : V0;

        v2f a;
        a.x = r0 * v0;
        a.y = r1 * v1;

        // c += A(16x4) * ones(4x16): accumulates row sums in f32
        c = __builtin_amdgcn_wmma_f32_16x16x4_f32(
                /*neg_a=*/false, a,
                /*neg_b=*/false, ones,
                /*c_mod=*/(short)0, c,
                /*reuse_a=*/false, /*reuse_b=*/false);
    }

    // D layout: VGPR r, lanes 0-15 -> M=r; lanes 16-31 -> M=8+r.
    // All columns hold the same row sum, so lane 0 / lane 16 write 8 rows each.
    if (lane == 0) {
#pragma unroll
        for (int j = 0; j < 8; ++j)
            out[rowBase + j] = c[j] + bb2;
    } else if (lane == 16) {
#pragma unroll
        for (int j = 0; j < 8; ++j)
            out[rowBase + 8 + j] = c[j] + bb2;
    }
}

extern "C" void kernel_launch(void* const* d_in, const int* in_sizes, int n_in,
                              void* d_out, int out_size, void* d_ws, size_t ws_size,
                              hipStream_t stream) {
    const float* inputs = (const float*)d_in[0];
    const float* V      = (const float*)d_in[1];
    const float* W      = (const float*)d_in[2];
    const float* b      = (const float*)d_in[3];
    const float* b2     = (const float*)d_in[4];
    float* out = (float*)d_out;

    const long long totalRows = (long long)in_sizes[0] / ROW_STRIDE; // == out_size
    const long long tiles  = (totalRows + ROWS_PER_WAVE - 1) / ROWS_PER_WAVE;
    const long long blocks = (tiles + WAVES_PER_BLOCK - 1) / WAVES_PER_BLOCK;

    irv_wmma_kernel<<<(int)blocks, 256, 0, stream>>>(
        inputs, V, W, b, b2, out, totalRows);
}